// CBF_24172075942522
// MI455X (gfx1250) — compile-verified
//
#include <hip/hip_runtime.h>
#include <hip/hip_bf16.h>
#include <stdint.h>

#define NAGENTS 4096
#define TOPK    12
#define NCOL    (NAGENTS * TOPK)     // 49152 columns, 3072 tiles of 16
#define WPW     4                    // waves (tiles) per workgroup
#define FSTRIDE 96                   // padded per-agent feature record (K=8 x 12)
#define FEPS    1e-4f

typedef __attribute__((ext_vector_type(2))) float v2f;
typedef __attribute__((ext_vector_type(8))) float v8f;

// ---------------------------------------------------------------------------
// Kernel 1: one wave per agent — fused pairwise distance + top-12 + features
// ---------------------------------------------------------------------------
__global__ __launch_bounds__(128) void cbf_topk_feat(
    const float* __restrict__ states,   // [N,4]
    float* __restrict__ feat,           // ws: [N,8,12] (rows 0..5 = features)
    float* __restrict__ mask_out)       // [N,12]
{
    const int wave = threadIdx.x >> 5;
    const int lane = threadIdx.x & 31;
    const int i = blockIdx.x * 4 + wave;

    const float sx = states[i * 4 + 0];
    const float sy = states[i * 4 + 1];
    const float sz = states[i * 4 + 2];
    const float sw = states[i * 4 + 3];

    // per-lane sorted list of packed (dist_bits, index) keys, ascending
    unsigned long long best[TOPK];
#pragma unroll
    for (int p = 0; p < TOPK; ++p) best[p] = ~0ull;

    for (int t = 0; t < NAGENTS / 32; ++t) {
        const int j = t * 32 + lane;
        const float dx = sx - states[j * 4 + 0];
        const float dy = sy - states[j * 4 + 1];
        // faithful to reference: (dx^2+eps) + (dy^2+eps)
        const float d2 = (dx * dx + FEPS) + (dy * dy + FEPS);
        const float dist = sqrtf(d2);
        unsigned long long key =
            ((unsigned long long)__float_as_uint(dist) << 32) | (unsigned)j;
        if (key < best[TOPK - 1]) {
#pragma unroll
            for (int p = 0; p < TOPK; ++p) {   // register-only bubble insert
                unsigned long long lo = best[p] < key ? best[p] : key;
                key = best[p] < key ? key : best[p];
                best[p] = lo;
            }
        }
    }

    // merge 32 lane-local lists: 12 rounds of wave-min extraction
    unsigned long long mine = ~0ull;
    for (int r = 0; r < TOPK; ++r) {
        unsigned long long v = best[0];
#pragma unroll
        for (int off = 16; off >= 1; off >>= 1) {
            unsigned long long o = __shfl_xor(v, off, 32);
            v = o < v ? o : v;
        }
        if (best[0] == v) {                    // unique winner pops its head
#pragma unroll
            for (int p = 0; p < TOPK - 1; ++p) best[p] = best[p + 1];
            best[TOPK - 1] = ~0ull;
        }
        if (lane == r) mine = v;               // lane r keeps r-th nearest
    }

    if (lane < TOPK) {
        const unsigned j = (unsigned)(mine & 0xffffffffull);
        const float dist = __uint_as_float((unsigned)(mine >> 32));
        // flat [12][6] record == rows 0..5 of the [8][12] B-matrix (reshape!)
        float* dst = feat + (size_t)i * FSTRIDE + lane * 6;
        dst[0] = sx - states[j * 4 + 0];
        dst[1] = sy - states[j * 4 + 1];
        dst[2] = sz - states[j * 4 + 2];
        dst[3] = sw - states[j * 4 + 3];
        dst[4] = (j == (unsigned)i) ? 1.0f : 0.0f;
        dst[5] = dist - 0.8f;
        // zero-fill pad rows 6..7 (indices 72..95), two per lane
        feat[(size_t)i * FSTRIDE + 72 + lane * 2 + 0] = 0.0f;
        feat[(size_t)i * FSTRIDE + 72 + lane * 2 + 1] = 0.0f;
        mask_out[i * TOPK + lane] = (dist <= 1.0f) ? 1.0f : 0.0f;
    }
}

// ---------------------------------------------------------------------------
// Kernel 2: batched MLP via v_wmma_f32_16x16x4_f32, one wave per 16 columns
// ---------------------------------------------------------------------------
__global__ __launch_bounds__(128) void cbf_mlp_wmma(
    const float* __restrict__ feat,
    const float* __restrict__ W1, const float* __restrict__ b1,
    const float* __restrict__ W2, const float* __restrict__ b2,
    const float* __restrict__ W3, const float* __restrict__ b3,
    const float* __restrict__ W4, const float* __restrict__ b4,
    const float* __restrict__ mask, float* __restrict__ out)
{
    __shared__ float bufA[WPW][64 * 16];    // H1 / H3 stage (per-wave slice)
    __shared__ float bufB[WPW][128 * 16];   // H2 stage       (per-wave slice)
    __shared__ float w1pad[64 * 8];         // W1 zero-padded [64][8], WG-shared

    const int wave = threadIdx.x >> 5;
    const int lane = threadIdx.x & 31;
    const int tile = blockIdx.x * WPW + wave;
    const int n    = lane & 15;             // col within tile == A-row M%16
    const int hi   = lane >> 4;             // half-wave select
    const int col  = tile * 16 + n;         // global column (agent,slot)
    const int agent = col / TOPK;
    const int slot  = col % TOPK;
    float* hA = bufA[wave];
    float* hB = bufB[wave];

    // stage W1 zero-padded into LDS (once per workgroup)
#pragma unroll
    for (int e = 0; e < 4; ++e) {
        const int idx = threadIdx.x * 4 + e;       // 128*4 = 512 = 64*8
        const int m = idx >> 3, k = idx & 7;
        w1pad[idx] = (k < 6) ? W1[m * 6 + k] : 0.0f;
    }
    __syncthreads();

    // ---- layer 1: H1 = relu(W1[64x8pad] * F[8x16]) -------------------------
    const float* fb = feat + (size_t)agent * FSTRIDE + slot;  // F[k][slot], stride 12
    v2f bfr[2];
#pragma unroll
    for (int kc = 0; kc < 2; ++kc) {
        const int k0 = kc * 4 + 2 * hi;
        v2f b; b.x = fb[k0 * 12]; b.y = fb[(k0 + 1) * 12];    // unguarded
        bfr[kc] = b;
    }
#pragma unroll
    for (int mt = 0; mt < 4; ++mt) {
        v8f acc = {};
#pragma unroll
        for (int kc = 0; kc < 2; ++kc) {
            const int m  = mt * 16 + n;
            const int k0 = kc * 4 + 2 * hi;
            v2f a = *(const v2f*)(w1pad + m * 8 + k0);        // ds_load_b64
            acc = __builtin_amdgcn_wmma_f32_16x16x4_f32(
                false, a, false, bfr[kc], (short)0, acc, false, false);
        }
#pragma unroll
        for (int e = 0; e < 8; ++e) {
            const int mr = mt * 16 + 8 * hi + e;   // C-layout row
            const float x = acc[e] + b1[mr];
            hA[mr * 16 + n] = x > 0.0f ? x : 0.0f;
        }
    }
    __syncthreads();

    // ---- layer 2: H2 = relu(W2[128x64] * H1[64x16]) ------------------------
    // hoist B fragments (invariant across the 8 M-tiles): 16 x v2f = 32 VGPRs
    v2f b2f[16];
#pragma unroll
    for (int kc = 0; kc < 16; ++kc) {
        const int k0 = kc * 4 + 2 * hi;
        v2f b; b.x = hA[k0 * 16 + n]; b.y = hA[(k0 + 1) * 16 + n];
        b2f[kc] = b;
    }
    for (int mt = 0; mt < 8; ++mt) {
        v8f acc = {};
        const int m = mt * 16 + n;
#pragma unroll 8
        for (int kc = 0; kc < 16; ++kc) {
            const int k0 = kc * 4 + 2 * hi;
            v2f a = *(const v2f*)(W2 + m * 64 + k0);          // global_load_b64
            acc = __builtin_amdgcn_wmma_f32_16x16x4_f32(
                false, a, false, b2f[kc], (short)0, acc, false, false);
        }
#pragma unroll
        for (int e = 0; e < 8; ++e) {
            const int mr = mt * 16 + 8 * hi + e;
            const float x = acc[e] + b2[mr];
            hB[mr * 16 + n] = x > 0.0f ? x : 0.0f;
        }
    }
    __syncthreads();

    // ---- layer 3: H3 = relu(W3[64x128] * H2[128x16]) -----------------------
    for (int mt = 0; mt < 4; ++mt) {
        v8f acc = {};
        const int m = mt * 16 + n;
#pragma unroll 8
        for (int kc = 0; kc < 32; ++kc) {
            const int k0 = kc * 4 + 2 * hi;
            v2f a = *(const v2f*)(W3 + m * 128 + k0);
            v2f b; b.x = hB[k0 * 16 + n]; b.y = hB[(k0 + 1) * 16 + n];
            acc = __builtin_amdgcn_wmma_f32_16x16x4_f32(
                false, a, false, b, (short)0, acc, false, false);
        }
#pragma unroll
        for (int e = 0; e < 8; ++e) {
            const int mr = mt * 16 + 8 * hi + e;
            const float x = acc[e] + b3[mr];
            hA[mr * 16 + n] = x > 0.0f ? x : 0.0f;
        }
    }
    __syncthreads();

    // ---- layer 4: out = (W4[1x64] * H3 + b4) * mask ------------------------
    float part = 0.0f;
#pragma unroll 8
    for (int kk = 0; kk < 32; ++kk) {
        const int k = hi * 32 + kk;
        part += W4[k] * hA[k * 16 + n];
    }
    part += __shfl_xor(part, 16, 32);
    if (hi == 0) {
        out[col] = (part + b4[0]) * mask[col];
    }
}

// ---------------------------------------------------------------------------
extern "C" void kernel_launch(void* const* d_in, const int* in_sizes, int n_in,
                              void* d_out, int out_size, void* d_ws, size_t ws_size,
                              hipStream_t stream) {
    const float* states = (const float*)d_in[0];
    const float* W1 = (const float*)d_in[1];
    const float* b1 = (const float*)d_in[2];
    const float* W2 = (const float*)d_in[3];
    const float* b2 = (const float*)d_in[4];
    const float* W3 = (const float*)d_in[5];
    const float* b3 = (const float*)d_in[6];
    const float* W4 = (const float*)d_in[7];
    const float* b4 = (const float*)d_in[8];

    float* out  = (float*)d_out;            // h in [0, NCOL), mask in [NCOL, 2*NCOL)
    float* mask = out + NCOL;
    float* feat = (float*)d_ws;             // 4096*96 f32 = 1.5 MB scratch

    cbf_topk_feat<<<NAGENTS / 4, 128, 0, stream>>>(states, feat, mask);
    cbf_mlp_wmma<<<(NCOL / 16) / WPW, 128, 0, stream>>>(
        feat, W1, b1, W2, b2, W3, b3, W4, b4, mask, out);
}